// L_Depth_52871047414179
// MI455X (gfx1250) — compile-verified
//
#include <hip/hip_runtime.h>
#include <hip/hip_bf16.h>

// ---------------- problem constants ----------------
#define NB      32          // batch
#define HW      (512 * 512) // pixels per plane
#define NBINS   256
#define TPB     256         // threads per block (8 wave32)
#define TILE    1024        // pixels staged per tile (TPB * 4 via b128)
#define PPB     8192        // pixels per block
#define NITER   (PPB / TILE)           // 8
#define BLK_PER_BATCH (HW / PPB)       // 32

// ---------------- CDNA5 async global->LDS path ----------------
#if defined(__gfx1250__) && __has_builtin(__builtin_amdgcn_global_load_async_to_lds_b128)
#define ASYNC_LDS 1
#else
#define ASYNC_LDS 0
#endif

#if defined(__gfx1250__) && __has_builtin(__builtin_amdgcn_s_wait_asynccnt)
#define WAIT_BUILTIN 1
#else
#define WAIT_BUILTIN 0
#endif

typedef int v4i __attribute__((ext_vector_type(4)));

__device__ __forceinline__ void async_copy_b128(const void* gsrc, void* ldst) {
#if ASYNC_LDS
  // 16 bytes per lane, memory -> LDS, tracked by ASYNCcnt
  __builtin_amdgcn_global_load_async_to_lds_b128(
      (__attribute__((address_space(1))) v4i*)gsrc,
      (__attribute__((address_space(3))) v4i*)ldst, 0, 0);
#else
  *(float4*)ldst = *(const float4*)gsrc;
#endif
}

template <int N>
__device__ __forceinline__ void wait_async() {
#if WAIT_BUILTIN
  __builtin_amdgcn_s_wait_asynccnt(N);
#elif ASYNC_LDS
  asm volatile("s_wait_asynccnt %0" ::"i"(N) : "memory");
#endif
  asm volatile("" ::: "memory"); // keep LDS reads below the wait
}

// ---------------- main streaming histogram kernel (first in file => shows in disasm) ----------------
__global__ __launch_bounds__(TPB)
void hist_kernel(const float* __restrict__ img, const float* __restrict__ depth,
                 float* __restrict__ gsum, float* __restrict__ gcnt) {
  const int b = blockIdx.y;
  const int t = threadIdx.x;
  const int base = blockIdx.x * PPB;

  const float* __restrict__ R = img + (size_t)b * 3 * HW;
  const float* __restrict__ G = R + HW;
  const float* __restrict__ Bp = R + 2 * HW;
  const float* __restrict__ D = depth + (size_t)b * HW;

  // staging tiles: [buf][chan R,G,B,D][TILE] = 32 KB ; histogram = 2 KB
  __shared__ __align__(16) float tiles[2][4][TILE];
  __shared__ float    hsum[NBINS];
  __shared__ unsigned hcnt[NBINS];

  hsum[t] = 0.0f;
  hcnt[t] = 0u;
  __syncthreads();

  const int lane4 = t * 4; // this thread's 4-float slot within a tile

  // issue one tile (4 async b128 loads from this lane; lane consumes only its own slots)
  auto issue = [&](int it, int buf) {
    const int off = base + it * TILE + lane4;
    async_copy_b128(R  + off, &tiles[buf][0][lane4]);
    async_copy_b128(G  + off, &tiles[buf][1][lane4]);
    async_copy_b128(Bp + off, &tiles[buf][2][lane4]);
    async_copy_b128(D  + off, &tiles[buf][3][lane4]);
  };

  auto process = [&](int buf) {
    float4 r4 = *(const float4*)&tiles[buf][0][lane4];
    float4 g4 = *(const float4*)&tiles[buf][1][lane4];
    float4 b4 = *(const float4*)&tiles[buf][2][lane4];
    float4 d4 = *(const float4*)&tiles[buf][3][lane4];
    const float rr[4] = {r4.x, r4.y, r4.z, r4.w};
    const float gg[4] = {g4.x, g4.y, g4.z, g4.w};
    const float bb[4] = {b4.x, b4.y, b4.z, b4.w};
    const float dd[4] = {d4.x, d4.y, d4.z, d4.w};
#pragma unroll
    for (int j = 0; j < 4; ++j) {
      float illum = fmaf(0.2f, rr[j], fmaf(0.7f, gg[j], 0.1f * bb[j]));
      int idx = __float2int_rn(dd[j] * 255.0f);
      idx = idx < 0 ? 0 : (idx > 255 ? 255 : idx);
      atomicAdd(&hsum[idx], illum); // ds_add_f32
      atomicAdd(&hcnt[idx], 1u);    // ds_add_u32
    }
  };

  int buf = 0;
  issue(0, buf);
#pragma unroll
  for (int it = 0; it < NITER; ++it) {
    const int nbuf = buf ^ 1;
    if (it + 1 < NITER) {
      issue(it + 1, nbuf);
      wait_async<4>(); // current tile's 4 loads done; next tile's 4 still in flight
    } else {
      wait_async<0>();
    }
    process(buf);
    buf = nbuf;
  }

  __syncthreads();
  // flush block-private histogram to global (one bin per thread)
  atomicAdd(&gsum[b * NBINS + t], hsum[t]);
  atomicAdd(&gcnt[b * NBINS + t], (float)hcnt[t]);
}

// ---------------- helper kernels ----------------
__global__ void zero_ws_kernel(float* __restrict__ p, int n) {
  int i = blockIdx.x * blockDim.x + threadIdx.x;
  if (i < n) p[i] = 0.0f;
}

__global__ __launch_bounds__(TPB)
void finalize_kernel(const float* __restrict__ gsum, const float* __restrict__ gcnt,
                     float* __restrict__ out) {
  const int t = threadIdx.x;
  float acc = 0.0f;
  for (int i = t; i < NB * (NBINS - 1); i += TPB) {
    const int b = i / (NBINS - 1);
    const int k = i - b * (NBINS - 1);
    const float* s = gsum + b * NBINS;
    const float* c = gcnt + b * NBINS;
    const float m0 = s[k] / c[k];
    const float m1 = s[k + 1] / c[k + 1];
    acc += fmaxf(m0 - m1, 0.0f);
  }
  __shared__ float red[TPB];
  red[t] = acc;
  __syncthreads();
  for (int s = TPB / 2; s > 0; s >>= 1) {
    if (t < s) red[t] += red[t + s];
    __syncthreads();
  }
  if (t == 0) out[0] = red[0] / (255.0f * (float)NB);
}

// ---------------- launch ----------------
extern "C" void kernel_launch(void* const* d_in, const int* in_sizes, int n_in,
                              void* d_out, int out_size, void* d_ws, size_t ws_size,
                              hipStream_t stream) {
  (void)in_sizes; (void)n_in; (void)out_size; (void)ws_size;
  const float* img   = (const float*)d_in[0]; // [32,3,512,512] f32
  const float* depth = (const float*)d_in[1]; // [32,1,512,512] f32

  float* gsum = (float*)d_ws;           // [32,256]
  float* gcnt = gsum + NB * NBINS;      // [32,256]
  const int nws = 2 * NB * NBINS;       // 16384 floats

  zero_ws_kernel<<<(nws + TPB - 1) / TPB, TPB, 0, stream>>>(gsum, nws);

  dim3 grid(BLK_PER_BATCH, NB);
  hist_kernel<<<grid, TPB, 0, stream>>>(img, depth, gsum, gcnt);

  finalize_kernel<<<1, TPB, 0, stream>>>(gsum, gcnt, (float*)d_out);
}